// Net_31688268709989
// MI455X (gfx1250) — compile-verified
//
#include <hip/hip_runtime.h>
#include <hip/hip_bf16.h>

typedef __attribute__((ext_vector_type(2))) float v2f;
typedef __attribute__((ext_vector_type(8))) float v8f;

#define EMB 64

// ---------------------------------------------------------------------------
// Degree / normalization
// ---------------------------------------------------------------------------
__global__ void k_init_deg(float* deg, int n) {
    int i = blockIdx.x * blockDim.x + threadIdx.x;
    if (i < n) deg[i] = 1.0f;                      // self-loop contribution
}

__global__ void k_edge_deg(const int* __restrict__ dst, float* deg, int E) {
    int e = blockIdx.x * blockDim.x + threadIdx.x;
    if (e < E) atomicAdd(&deg[dst[e]], 1.0f);
}

__global__ void k_make_dinv(float* deg, int n) {
    int i = blockIdx.x * blockDim.x + threadIdx.x;
    if (i < n) {
        float d = deg[i];
        deg[i] = (d > 0.0f) ? rsqrtf(d) : 0.0f;
    }
}

// ---------------------------------------------------------------------------
// FP32 WMMA GEMM: out[M x 64] = A[M x K] @ W[K x 64]
// One wave computes a 16-row x 64-col strip using 4 accumulators and
// V_WMMA_F32_16X16X4_F32 (K advances 4 per instruction).
//
// A 16x4 f32 layout: lanes 0-15 hold M=0..15; VGPR0/1 = K={0,1} for lanes
// 0-15 and K={2,3} for lanes 16-31  -> per lane: m = lane&15, kh = (lane>>4)*2,
// a = {A[m][kb+kh], A[m][kb+kh+1]}  (contiguous -> one 8B load).
// B 4x16 f32 layout (row striped across lanes): per lane: n = lane&15,
// b = {W[kb+kh][n], W[kb+kh+1][n]}.
// C/D 16x16 f32: element r of v8f = D[r + 8*(lane>>4)][lane&15].
// ---------------------------------------------------------------------------
template <int K>
__global__ void k_gemm_wmma(const float* __restrict__ A,
                            const float* __restrict__ W,
                            float* __restrict__ out, int M) {
    __shared__ float ldsW[K * EMB];

    // Stage W (K x 64 fp32, <=32KB) into LDS, all threads participate.
    {
        const float4* Wv = (const float4*)W;
        float4* Lv = (float4*)ldsW;
        for (int i = threadIdx.x; i < K * (EMB / 4); i += blockDim.x)
            Lv[i] = Wv[i];
    }
    __syncthreads();

    const int lane = threadIdx.x & 31;
    const int wid  = threadIdx.x >> 5;
    const int row_base = (blockIdx.x * (int)(blockDim.x >> 5) + wid) * 16;
    if (row_base >= M) return;                 // wave-uniform: EXEC stays all-1s

    const int m    = row_base + (lane & 15);
    const int kh   = (lane >> 4) << 1;         // 0 or 2
    const int col0 = lane & 15;

    v8f acc0 = {}, acc1 = {}, acc2 = {}, acc3 = {};
    const float* arow = A + (size_t)m * K;

#pragma unroll 4
    for (int kb = 0; kb < K; kb += 4) {
        v2f a = *(const v2f*)(arow + kb + kh);

        const float* w0 = ldsW + (kb + kh) * EMB + col0;
        const float* w1 = w0 + EMB;

        v2f b;
        b[0] = w0[0];  b[1] = w1[0];
        acc0 = __builtin_amdgcn_wmma_f32_16x16x4_f32(false, a, false, b,
                                                     (short)0, acc0, false, false);
        b[0] = w0[16]; b[1] = w1[16];
        acc1 = __builtin_amdgcn_wmma_f32_16x16x4_f32(false, a, false, b,
                                                     (short)0, acc1, false, false);
        b[0] = w0[32]; b[1] = w1[32];
        acc2 = __builtin_amdgcn_wmma_f32_16x16x4_f32(false, a, false, b,
                                                     (short)0, acc2, false, false);
        b[0] = w0[48]; b[1] = w1[48];
        acc3 = __builtin_amdgcn_wmma_f32_16x16x4_f32(false, a, false, b,
                                                     (short)0, acc3, false, false);
    }

    const int rhalf = (lane >> 4) * 8;
#pragma unroll
    for (int r = 0; r < 8; ++r) {
        float* orow = out + (size_t)(row_base + r + rhalf) * EMB + col0;
        orow[0]  = acc0[r];
        orow[16] = acc1[r];
        orow[32] = acc2[r];
        orow[48] = acc3[r];
    }
}

// ---------------------------------------------------------------------------
// Aggregation: agg = b + dinv[i]^2 * h[i]  (self-loop + bias), then per-edge
// atomic scatter-add of dinv[s]*dinv[d]*h[s], then (optionally) ReLU.
// ---------------------------------------------------------------------------
__global__ void k_init_agg(const float* __restrict__ h,
                           const float* __restrict__ dinv,
                           const float* __restrict__ bias,
                           float* __restrict__ agg, int n) {
    int t = blockIdx.x * blockDim.x + threadIdx.x;       // over n*16 float4s
    if (t >= n * 16) return;
    int node = t >> 4;
    int f4   = (t & 15) * 4;
    float di = dinv[node];
    float w  = di * di;
    float4 hv = *(const float4*)(h + (size_t)node * EMB + f4);
    float4 bv = *(const float4*)(bias + f4);
    float4 o;
    o.x = bv.x + hv.x * w;
    o.y = bv.y + hv.y * w;
    o.z = bv.z + hv.z * w;
    o.w = bv.w + hv.w * w;
    *(float4*)(agg + (size_t)node * EMB + f4) = o;
}

__global__ void k_edge_agg(const int* __restrict__ src,
                           const int* __restrict__ dst,
                           const float* __restrict__ dinv,
                           const float* __restrict__ h,
                           float* __restrict__ agg, int E) {
    long long t = (long long)blockIdx.x * blockDim.x + threadIdx.x; // E*16
    if (t >= (long long)E * 16) return;
    int e  = (int)(t >> 4);
    int f4 = ((int)t & 15) * 4;
    int s = src[e];
    int d = dst[e];
    float w = dinv[s] * dinv[d];
    float4 hv = *(const float4*)(h + (size_t)s * EMB + f4);
    float* o = agg + (size_t)d * EMB + f4;
    atomicAdd(o + 0, hv.x * w);
    atomicAdd(o + 1, hv.y * w);
    atomicAdd(o + 2, hv.z * w);
    atomicAdd(o + 3, hv.w * w);
}

__global__ void k_relu(float* __restrict__ agg, int n) {
    int t = blockIdx.x * blockDim.x + threadIdx.x;       // over n*16 float4s
    if (t >= n * 16) return;
    float4* p = ((float4*)agg) + t;
    float4 v = *p;
    v.x = fmaxf(v.x, 0.0f);
    v.y = fmaxf(v.y, 0.0f);
    v.z = fmaxf(v.z, 0.0f);
    v.w = fmaxf(v.w, 0.0f);
    *p = v;
}

// ---------------------------------------------------------------------------
// Head with fused ReLU: out[i] = dot(max(h[i,:],0), W3) + b3
// (the second GCN layer's ReLU output feeds only the head, so we never
//  materialize it -- saves a full 2x25.6MB pass)
// ---------------------------------------------------------------------------
__global__ void k_head_relu(const float* __restrict__ h,
                            const float* __restrict__ W3,
                            const float* __restrict__ b3,
                            float* __restrict__ out, int n) {
    int i = blockIdx.x * blockDim.x + threadIdx.x;
    if (i >= n) return;
    const float4* hp = (const float4*)(h + (size_t)i * EMB);
    const float4* wp = (const float4*)W3;
    float acc = 0.0f;
#pragma unroll
    for (int j = 0; j < EMB / 4; ++j) {
        float4 hv = hp[j];
        float4 wv = wp[j];
        acc += fmaxf(hv.x, 0.0f) * wv.x + fmaxf(hv.y, 0.0f) * wv.y +
               fmaxf(hv.z, 0.0f) * wv.z + fmaxf(hv.w, 0.0f) * wv.w;
    }
    out[i] = acc + b3[0];
}

// ---------------------------------------------------------------------------
extern "C" void kernel_launch(void* const* d_in, const int* in_sizes, int n_in,
                              void* d_out, int out_size, void* d_ws, size_t ws_size,
                              hipStream_t stream) {
    const float* x  = (const float*)d_in[0];
    const int*   ei = (const int*)  d_in[1];
    const float* W1 = (const float*)d_in[2];
    const float* b1 = (const float*)d_in[3];
    const float* W2 = (const float*)d_in[4];
    const float* b2 = (const float*)d_in[5];
    const float* W3 = (const float*)d_in[6];
    const float* b3 = (const float*)d_in[7];
    float* out = (float*)d_out;

    const int N = in_sizes[0] / 128;   // 100000
    const int E = in_sizes[1] / 2;     // 1600000
    const int* src = ei;
    const int* dst = ei + E;

    // Workspace layout: dinv[N] | bufA[N*64] | bufB[N*64]
    char* ws = (char*)d_ws;
    float* dinv = (float*)ws;
    size_t offA = ((size_t)N * sizeof(float) + 255) & ~(size_t)255;
    float* bufA = (float*)(ws + offA);
    size_t offB = offA + (((size_t)N * EMB * sizeof(float) + 255) & ~(size_t)255);
    float* bufB = (float*)(ws + offB);

    const int T = 256;
    const int nb_node  = (N + T - 1) / T;
    const int nb_edge  = (E + T - 1) / T;
    const int nb_feat  = (N * 16 + T - 1) / T;                       // N*64/4
    const int nb_eagg  = (int)(((long long)E * 16 + T - 1) / T);
    const int nb_gemm  = (N + 127) / 128;                            // 8 waves * 16 rows

    // Symmetric normalization coefficients
    k_init_deg <<<nb_node, T, 0, stream>>>(dinv, N);
    k_edge_deg <<<nb_edge, T, 0, stream>>>(dst, dinv, E);
    k_make_dinv<<<nb_node, T, 0, stream>>>(dinv, N);

    // Layer 1: h = relu(aggregate(x @ W1) + b1)
    k_gemm_wmma<128><<<nb_gemm, T, 0, stream>>>(x, W1, bufA, N);
    k_init_agg <<<nb_feat, T, 0, stream>>>(bufA, dinv, b1, bufB, N);
    k_edge_agg <<<nb_eagg, T, 0, stream>>>(src, dst, dinv, bufA, bufB, E);
    k_relu     <<<nb_feat, T, 0, stream>>>(bufB, N);

    // Layer 2 (pre-activation): aggregate(h @ W2) + b2
    k_gemm_wmma<64><<<nb_gemm, T, 0, stream>>>(bufB, W2, bufA, N);
    k_init_agg <<<nb_feat, T, 0, stream>>>(bufA, dinv, b2, bufB, N);
    k_edge_agg <<<nb_eagg, T, 0, stream>>>(src, dst, dinv, bufA, bufB, E);

    // Head with fused final ReLU
    k_head_relu<<<nb_node, T, 0, stream>>>(bufB, W3, b3, out, N);
}